// VNLoss_34505767256605
// MI455X (gfx1250) — compile-verified
//
#include <hip/hip_runtime.h>
#include <hip/hip_bf16.h>

#define BB 8
#define HH 512
#define WW 1024
#define GG 104857
#define NTRIP (BB * GG)      // 838856 triplets
#define HWSZ (HH * WW)
#define EPSV 1e-6f
#define DCOS 0.867f
#define DDIFF 0.005f
#define DZPAD 1e-5f

typedef __attribute__((ext_vector_type(2))) float v2f;
typedef __attribute__((ext_vector_type(8))) float v8f;

struct F3 { float x, y, z; };

__device__ __forceinline__ float dot3(const F3& a, const F3& b) {
    return a.x * b.x + a.y * b.y + a.z * b.z;
}
__device__ __forceinline__ F3 sub3(const F3& a, const F3& b) {
    return {a.x - b.x, a.y - b.y, a.z - b.z};
}
__device__ __forceinline__ F3 cross3(const F3& a, const F3& b) {
    return {a.y * b.z - a.z * b.y,
            a.z * b.x - a.x * b.z,
            a.x * b.y - a.y * b.x};
}
__device__ __forceinline__ F3 tri_normal(const F3& p0, const F3& p1, const F3& p2) {
    F3 n = cross3(sub3(p1, p0), sub3(p2, p0));
    float nn = sqrtf(dot3(n, n));
    nn = nn + ((nn == 0.0f) ? EPSV : 0.0f);
    float inv = 1.0f / nn;
    return {n.x * inv, n.y * inv, n.z * inv};
}

// Full-wave (32 lane) sum in ONE v_wmma_f32_16x16x4_f32:
// A (16x4): vgpr0 = lane value (lanes0-15 -> (M=lane,K=0), lanes16-31 -> (M=lane-16,K=2)),
// vgpr1 = 0. B (4x16) = ones. D[m][n] = rowsum_A(m). Per-lane sum of the 8 D VGPRs gives
// rows 0-7 (lanes<16) / rows 8-15 (lanes>=16); readlane(0)+readlane(16) = wave total.
// Must be called with full EXEC (all lanes converged).
__device__ __forceinline__ float wave_sum_wmma(float v) {
    v2f a; a[0] = v;    a[1] = 0.0f;
    v2f b; b[0] = 1.0f; b[1] = 1.0f;
    v8f c = {0.f, 0.f, 0.f, 0.f, 0.f, 0.f, 0.f, 0.f};
    c = __builtin_amdgcn_wmma_f32_16x16x4_f32(false, a, false, b, (short)0, c, false, false);
    float s = c[0] + c[1] + c[2] + c[3] + c[4] + c[5] + c[6] + c[7];
    int sb = __float_as_int(s);
    float lo = __int_as_float(__builtin_amdgcn_readlane(sb, 0));
    float hi = __int_as_float(__builtin_amdgcn_readlane(sb, 16));
    return lo + hi;
}

// ------------------------------------------------------------------ kernels

__global__ void vnl_zero(unsigned* __restrict__ p, int n) {
    int i = blockIdx.x * blockDim.x + threadIdx.x;
    if (i < n) p[i] = 0u;
}

// scalars layout (in ws): [0]=sum_valid f32, [1]=cnt_valid f32, [2]=k u32,
// [3]=bin1 u32, [4]=c_before u32, [5]=t_bits u32, [6]=c_lt u32, [7]=sum_lt f32
__global__ void vnl_main(const float* __restrict__ pred, const float* __restrict__ targ,
                         const float* __restrict__ intr,
                         const int* __restrict__ p1, const int* __restrict__ p2,
                         const int* __restrict__ p3,
                         unsigned* __restrict__ loss_bits, unsigned* __restrict__ hist1,
                         float* __restrict__ scal_f) {
    float lsum = 0.0f, lcnt = 0.0f;
    const int stride = gridDim.x * blockDim.x;
    for (int t = blockIdx.x * blockDim.x + threadIdx.x; t < NTRIP; t += stride) {
        const int b = t / GG;
        const float* tb = targ + (size_t)b * HWSZ;
        const float* pb = pred + (size_t)b * HWSZ;
        const float fx = intr[b * 9 + 0];
        const float u0 = intr[b * 9 + 2];
        const float v0 = intr[b * 9 + 5];
        int q[3];
        q[0] = p1[t]; q[1] = p2[t]; q[2] = p3[t];

        F3 gt[3], pr[3];
        #pragma unroll
        for (int i = 0; i < 3; ++i) {
            const int qq = q[i];
            const int v = qq / WW;
            const int u = qq - v * WW;
            const float du = (float)u - u0;
            const float dv = (float)v - v0;
            const float dz = tb[qq];                 // gathered GT depth
            const float pz = pb[qq];                 // gathered pred depth
            gt[i] = {du * dz / fx, dv * dz / fx, dz};
            pr[i] = {du * pz / fx, dv * pz / fx, pz};
        }

        // --- filter mask on GT triplet ---
        F3 dd[3];
        dd[0] = sub3(gt[1], gt[0]);   // 12
        dd[1] = sub3(gt[2], gt[0]);   // 13
        dd[2] = sub3(gt[2], gt[1]);   // 23
        float qn[3];
        #pragma unroll
        for (int i = 0; i < 3; ++i) qn[i] = sqrtf(dot3(dd[i], dd[i]));
        int cc = 0;
        #pragma unroll
        for (int i = 0; i < 3; ++i)
            #pragma unroll
            for (int j = 0; j < 3; ++j) {
                const float e = dot3(dd[i], dd[j]);
                const float ne = e / (qn[i] * qn[j] + EPSV);
                cc += (fabsf(ne) > DCOS) ? 1 : 0;
            }
        const bool mcos = cc > 3;
        const bool mpad = (gt[0].z > DZPAD) && (gt[1].z > DZPAD) && (gt[2].z > DZPAD);
        const bool mx = (fabsf(dd[0].x) < DDIFF) || (fabsf(dd[1].x) < DDIFF) || (fabsf(dd[2].x) < DDIFF);
        const bool my = (fabsf(dd[0].y) < DDIFF) || (fabsf(dd[1].y) < DDIFF) || (fabsf(dd[2].y) < DDIFF);
        const bool mz = (fabsf(dd[0].z) < DDIFF) || (fabsf(dd[1].z) < DDIFF) || (fabsf(dd[2].z) < DDIFF);
        const bool valid = mpad && !((mx && my && mz) || mcos);

        // faithful replication of the reference's pred z==0 patch
        // (cond over POINT j selects COORD row j of all points)
        if (pr[0].z == 0.0f) { pr[0].x = 1e-4f; pr[1].x = 1e-4f; pr[2].x = 1e-4f; }
        if (pr[1].z == 0.0f) { pr[0].y = 1e-4f; pr[1].y = 1e-4f; pr[2].y = 1e-4f; }
        if (pr[2].z == 0.0f) { pr[0].z = 1e-4f; pr[1].z = 1e-4f; pr[2].z = 1e-4f; }

        const F3 ng = tri_normal(gt[0], gt[1], gt[2]);
        const F3 np = tri_normal(pr[0], pr[1], pr[2]);
        const float loss = fabsf(ng.x - np.x) + fabsf(ng.y - np.y) + fabsf(ng.z - np.z);

        const unsigned bits = valid ? __float_as_uint(loss) : 0xFFFFFFFFu;
        loss_bits[t] = bits;
        if (valid) {
            atomicAdd(&hist1[bits >> 16], 1u);
            lsum += loss;
            lcnt += 1.0f;
        }
    }
    const float wsum = wave_sum_wmma(lsum);   // full EXEC here (grid-stride loop exited)
    const float wcnt = wave_sum_wmma(lcnt);
    if ((threadIdx.x & 31) == 0) {
        atomicAdd(&scal_f[0], wsum);
        atomicAdd(&scal_f[1], wcnt);
    }
}

__global__ void vnl_scan1(const unsigned* __restrict__ hist1,
                          float* __restrict__ scal_f, unsigned* __restrict__ scal_u) {
    const unsigned n = (unsigned)(scal_f[1] + 0.5f);
    const unsigned k = n / 4u;
    scal_u[2] = k;
    unsigned b1 = 0xFFFFFFFFu, cbefore = 0u;
    if (k > 0u) {
        unsigned cum = 0u;
        for (int i = 0; i < 65536; ++i) {
            const unsigned c = hist1[i];
            if (cum + c >= k) { b1 = (unsigned)i; cbefore = cum; break; }
            cum += c;
        }
    }
    scal_u[3] = b1;
    scal_u[4] = cbefore;
}

__global__ void vnl_hist2(const unsigned* __restrict__ loss_bits,
                          const unsigned* __restrict__ scal_u,
                          unsigned* __restrict__ hist2) {
    const unsigned b1 = scal_u[3];     // valid bins are <= 0x7FFF; sentinel never matches
    const int stride = gridDim.x * blockDim.x;
    for (int t = blockIdx.x * blockDim.x + threadIdx.x; t < NTRIP; t += stride) {
        const unsigned bits = loss_bits[t];
        if ((bits >> 16) == b1) atomicAdd(&hist2[bits & 0xFFFFu], 1u);
    }
}

__global__ void vnl_scan2(const unsigned* __restrict__ hist2, unsigned* __restrict__ scal_u) {
    const unsigned k = scal_u[2];
    unsigned t_bits = 0u, c_lt = 0u;
    if (k > 0u) {
        const unsigned b1 = scal_u[3];
        unsigned cum = scal_u[4];
        for (int l = 0; l < 65536; ++l) {
            const unsigned c = hist2[l];
            if (cum + c >= k) { t_bits = (b1 << 16) | (unsigned)l; c_lt = cum; break; }
            cum += c;
        }
    }
    scal_u[5] = t_bits;
    scal_u[6] = c_lt;
}

__global__ void vnl_sumlt(const unsigned* __restrict__ loss_bits,
                          const unsigned* __restrict__ scal_u,
                          float* __restrict__ scal_f) {
    const unsigned tb = scal_u[5];     // sentinel (MSB set) always >= tb -> auto-excluded
    float ls = 0.0f;
    const int stride = gridDim.x * blockDim.x;
    for (int t = blockIdx.x * blockDim.x + threadIdx.x; t < NTRIP; t += stride) {
        const unsigned bits = loss_bits[t];
        if (bits < tb) ls += __uint_as_float(bits);
    }
    const float w = wave_sum_wmma(ls);
    if ((threadIdx.x & 31) == 0) atomicAdd(&scal_f[7], w);
}

__global__ void vnl_final(const float* __restrict__ scal_f,
                          const unsigned* __restrict__ scal_u,
                          float* __restrict__ out) {
    const float sumv  = scal_f[0];
    const float sumlt = scal_f[7];
    const unsigned n    = (unsigned)(scal_f[1] + 0.5f);
    const unsigned k    = scal_u[2];
    const unsigned c_lt = scal_u[6];
    const float T = __uint_as_float(scal_u[5]);
    const float kept = sumv - sumlt - (float)(k - c_lt) * T;
    const unsigned d = (n > k) ? (n - k) : 1u;
    out[0] = kept / (float)d;   // SCALE == 1.0
}

// ------------------------------------------------------------------ launch

extern "C" void kernel_launch(void* const* d_in, const int* in_sizes, int n_in,
                              void* d_out, int out_size, void* d_ws, size_t ws_size,
                              hipStream_t stream) {
    (void)in_sizes; (void)n_in; (void)out_size; (void)ws_size;
    const float* prediction = (const float*)d_in[0];
    const float* target     = (const float*)d_in[1];
    // d_in[2] = mask (only used in host-side index selection; unused here)
    const float* intrinsic  = (const float*)d_in[3];
    const int*   p1         = (const int*)d_in[4];
    const int*   p2         = (const int*)d_in[5];
    const int*   p3         = (const int*)d_in[6];
    float* out = (float*)d_out;

    unsigned char* ws = (unsigned char*)d_ws;
    unsigned* loss_bits = (unsigned*)ws;                               // NTRIP u32 (3.36 MB)
    const size_t off = (((size_t)NTRIP * 4u) + 255u) & ~(size_t)255u;
    unsigned* hist1 = (unsigned*)(ws + off);                           // 65536 u32
    unsigned* hist2 = hist1 + 65536;                                   // 65536 u32
    float*    scal_f = (float*)(hist2 + 65536);                        // 8 dwords
    unsigned* scal_u = (unsigned*)scal_f;

    const int ZN = 65536 * 2 + 8;
    vnl_zero<<<(ZN + 255) / 256, 256, 0, stream>>>(hist1, ZN);
    vnl_main<<<1024, 256, 0, stream>>>(prediction, target, intrinsic, p1, p2, p3,
                                       loss_bits, hist1, scal_f);
    vnl_scan1<<<1, 1, 0, stream>>>(hist1, scal_f, scal_u);
    vnl_hist2<<<1024, 256, 0, stream>>>(loss_bits, scal_u, hist2);
    vnl_scan2<<<1, 1, 0, stream>>>(hist2, scal_u);
    vnl_sumlt<<<1024, 256, 0, stream>>>(loss_bits, scal_u, scal_f);
    vnl_final<<<1, 1, 0, stream>>>(scal_f, scal_u, out);
}